// GATEncoder_4217657885347
// MI455X (gfx1250) — compile-verified
//
#include <hip/hip_runtime.h>
#include <math.h>
#include <stdint.h>

// ---------------------------------------------------------------------------
// GAT encoder for MI455X (gfx1250). All matrix math via V_WMMA_F32_16X16X4_F32
// (exact f32 — whole problem is ~35MB, fits in the 192MB L2; latency-bound, so
// no reason to quantize). Attention hp-tiles staged with the CDNA5 async
// global->LDS DMA path (ASYNCcnt) instead of a VGPR round-trip.
//   F_OBS=8 frames, 16 scenes x 256 peds (4096 total), D_IN=32, D_HID=16,
//   H0=4 heads, D_OUT=32. 128 independent (frame,scene) problems.
// ---------------------------------------------------------------------------

typedef float v2f __attribute__((ext_vector_type(2)));
typedef float v8f __attribute__((ext_vector_type(8)));

#define NBATCH 128   /* F_OBS * N_SCENES */
#define NPED   256

// ---------------------------------------------------------------------------
// Kernel 1: relation matrix -> bitmask (8 u32 per row). One block per
// (frame,scene), one thread per row i.
// ---------------------------------------------------------------------------
__global__ void rel_kernel(const float* __restrict__ dire, unsigned* __restrict__ relb) {
    const int b = blockIdx.x;          // b = f*16 + scene  ->  row base = b*256
    const int i = threadIdx.x;
    __shared__ float px_s[NPED], py_s[NPED];

    const size_t base = ((size_t)b * NPED + i) * 6;   // obs_dire has 6 features
    const float px = dire[base + 2];
    const float py = dire[base + 3];
    const float hd = dire[base + 5];
    px_s[i] = px; py_s[i] = py;
    __syncthreads();

    const float up   = hd + 62.0f;
    const float down = hd - 62.0f;
    const bool  g360 = up > 360.0f;
    const bool  band = (up >= 62.0f) && (up <= 124.0f);

    unsigned word = 0u;
    for (int j = 0; j < NPED; ++j) {
        const float dx = px_s[j] - px;
        const float dy = py_s[j] - py;
        const float dist = sqrtf(dx * dx + dy * dy);
        const bool dmask = dist <= 156.0f;
        float ang = atan2f(dy, dx) * 57.29577951308232f;   // degrees
        if (ang < 0.0f) ang += 360.0f;
        const bool c1 = g360 && ((ang >= down && ang <= 360.0f) ||
                                 (ang >= 0.0f && ang <= up - 360.0f));
        const bool c2 = band && ((ang >= down + 360.0f && ang <= 360.0f) ||
                                 (ang >= 0.0f && ang <= up));
        const bool c3 = (!g360) && (!band) && (ang >= 0.0f && ang <= 360.0f);
        const bool r  = dmask && (c1 || c2 || c3);
        word |= (unsigned)r << (j & 31);
        if ((j & 31) == 31) {
            relb[((size_t)b * NPED + i) * 8 + (j >> 5)] = word;
            word = 0u;
        }
    }
}

// ---------------------------------------------------------------------------
// Kernel 2: instance norm over the 256 peds of one (frame,scene), per feature.
// One block per batch; wave w owns dims w, w+8, ...; lanes stride rows.
// ---------------------------------------------------------------------------
template <int D>
__global__ void inorm_kernel(const float* __restrict__ x, float* __restrict__ y) {
    const int b = blockIdx.x;
    const float* xb = x + (size_t)b * NPED * D;
    float*       yb = y + (size_t)b * NPED * D;
    __shared__ float mu_s[D], rs_s[D];

    const int tid = threadIdx.x, lane = tid & 31, w = tid >> 5;
    for (int d = w; d < D; d += 8) {
        float s = 0.0f, q = 0.0f;
        for (int r = lane; r < NPED; r += 32) {
            const float v = xb[r * D + d];
            s += v; q += v * v;
        }
        for (int o = 16; o >= 1; o >>= 1) {
            s += __shfl_down(s, o, 32);
            q += __shfl_down(q, o, 32);
        }
        if (lane == 0) {
            const float mu  = s * (1.0f / 256.0f);
            const float var = q * (1.0f / 256.0f) - mu * mu;
            mu_s[d] = mu;
            rs_s[d] = 1.0f / sqrtf(var + 1e-5f);
        }
    }
    __syncthreads();
    for (int i = tid; i < NPED * D; i += 256) {
        const int d = i % D;
        yb[i] = (xb[i] - mu_s[d]) * rs_s[d];
    }
}

// ---------------------------------------------------------------------------
// Kernel 3: batched GEMM  C[b] = A[b](256xK) @ W(KxN)  via WMMA f32 16x16x4.
// HEADED: W comes from w0 of shape (4 heads, K, 16) concatenated along N.
// One block (8 waves) per batch; each wave owns 16*N/16/8 output tiles.
// ---------------------------------------------------------------------------
template <int K, int N, bool HEADED>
__global__ void gemm_wmma_kernel(const float* __restrict__ A,
                                 const float* __restrict__ W,
                                 float* __restrict__ C) {
    const int b = blockIdx.x;
    const float* Ab = A + (size_t)b * NPED * K;
    float*       Cb = C + (size_t)b * NPED * N;
    const int lane = threadIdx.x & 31, w = threadIdx.x >> 5;
    const int m  = lane & 15;       // A row / B col within tile (ISA layout)
    const int h2 = lane >> 4;       // half-wave selects K sub-pair

    constexpr int NT = N / 16;
    for (int t = w; t < 16 * NT; t += 8) {
        const int mt = t & 15;
        const int nt = t >> 4;
        v8f c = {0.f,0.f,0.f,0.f,0.f,0.f,0.f,0.f};
        for (int k0 = 0; k0 < K; k0 += 4) {
            const int ka = k0 + 2 * h2;
            v2f av, bv;
            av.x = Ab[(mt * 16 + m) * K + ka];
            av.y = Ab[(mt * 16 + m) * K + ka + 1];
            const int n = nt * 16 + m;
            if (HEADED) {   // w0: [head][K][16], head = n>>4, e = n&15
                bv.x = W[(n >> 4) * (K * 16) + ka * 16 + (n & 15)];
                bv.y = W[(n >> 4) * (K * 16) + (ka + 1) * 16 + (n & 15)];
            } else {        // w1: plain [K][N]
                bv.x = W[ka * N + n];
                bv.y = W[(ka + 1) * N + n];
            }
            c = __builtin_amdgcn_wmma_f32_16x16x4_f32(
                    false, av, false, bv, (short)0, c, false, false);
        }
        #pragma unroll
        for (int r = 0; r < 8; ++r)
            Cb[(mt * 16 + r + 8 * h2) * N + nt * 16 + m] = c[r];
    }
}

// ---------------------------------------------------------------------------
// Kernel 4: src/dst attention logits: src[b,h,n] = hp[b,n,h,:] . a_src[h,:]
// ---------------------------------------------------------------------------
template <int HEADS, int E>
__global__ void srcdst_kernel(const float* __restrict__ hp,
                              const float* __restrict__ a_src,
                              const float* __restrict__ a_dst,
                              float* __restrict__ srcv, float* __restrict__ dstv) {
    const int b = blockIdx.x, n = threadIdx.x;
    const float* row = hp + ((size_t)b * NPED + n) * (HEADS * E);
    #pragma unroll
    for (int h = 0; h < HEADS; ++h) {
        float s = 0.0f, d = 0.0f;
        #pragma unroll
        for (int e = 0; e < E; ++e) {
            const float v = row[h * E + e];
            s += v * a_src[h * E + e];
            d += v * a_dst[h * E + e];
        }
        srcv[((size_t)b * HEADS + h) * NPED + n] = s;
        dstv[((size_t)b * HEADS + h) * NPED + n] = d;
    }
}

// ---------------------------------------------------------------------------
// Kernel 5: fused masked-attention + softmax + (P @ hp) via WMMA, plus the
// layer epilogue (bias [+ELU] for layer0, bias only for layer1).
// Reference semantics: masked entries enter softmax as exp(0), NOT -inf.
// Half-lane pair (l, l+16) jointly owns one row; stats merged via shfl_xor 16.
// hp tile staged global->LDS with the CDNA5 async DMA path (ASYNCcnt).
// ---------------------------------------------------------------------------
template <int HEADS, int E, bool LAYER0>
__global__ void attn_kernel(const float* __restrict__ hp,
                            const float* __restrict__ srcv,
                            const float* __restrict__ dstv,
                            const unsigned* __restrict__ relb,
                            const float* __restrict__ bias,
                            float* __restrict__ out) {
    constexpr int OUTD = HEADS * E;
    constexpr int NT   = E / 16;
    const int b = blockIdx.x;
    const size_t gbase = (size_t)b * NPED;

    __shared__ __align__(16) float hp_s[NPED * E];
    __shared__ __align__(16) float dst_s[NPED];

    const int tid = threadIdx.x, lane = tid & 31, w = tid >> 5;
    const int m = lane & 15, h2 = lane >> 4;

    for (int h = 0; h < HEADS; ++h) {
        __syncthreads();   // previous head's LDS consumers are done

        // ---- async global->LDS staging (no VGPR round-trip, ASYNCcnt) ----
        // hp rows are OUTD-strided; each 4-float chunk is 16B aligned.
        for (int i = tid * 4; i < NPED * E; i += 256 * 4) {
            const int n = i / E, e = i % E;
            const float* gp = &hp[(gbase + n) * OUTD + h * E + e];
            const unsigned lo = (unsigned)(uintptr_t)(&hp_s[i]);
            asm volatile("global_load_async_to_lds_b128 %0, %1, off"
                         :: "v"(lo), "v"(gp) : "memory");
        }
        {
            const float* gp = &dstv[((size_t)b * HEADS + h) * NPED + tid];
            const unsigned lo = (unsigned)(uintptr_t)(&dst_s[tid]);
            asm volatile("global_load_async_to_lds_b32 %0, %1, off"
                         :: "v"(lo), "v"(gp) : "memory");
        }
        asm volatile("s_wait_asynccnt 0x0" ::: "memory");
        __syncthreads();

        for (int mt = w; mt < 16; mt += 8) {
            const int i = mt * 16 + m;                       // my row
            const float si = srcv[((size_t)b * HEADS + h) * NPED + i];
            unsigned rw[8];
            #pragma unroll
            for (int q = 0; q < 8; ++q)
                rw[q] = relb[(gbase + i) * 8 + q];

            // pass 1: row max (masked entries contribute 0)
            float mx = -INFINITY;
            #pragma unroll
            for (int q = 0; q < 8; ++q) {
                const unsigned wd = rw[q];
                for (int tt = 0; tt < 8; ++tt) {
                    const int j0 = (q * 8 + tt) * 4 + 2 * h2;
                    #pragma unroll
                    for (int u = 0; u < 2; ++u) {
                        const int j = j0 + u;
                        const float z = si + dst_s[j];
                        const float lv = z >= 0.0f ? z : 0.2f * z;
                        const float v = ((wd >> (j & 31)) & 1u) ? lv : 0.0f;
                        mx = fmaxf(mx, v);
                    }
                }
            }
            mx = fmaxf(mx, __shfl_xor(mx, 16, 32));

            // pass 2: row sum of exp(v - mx) over ALL 256 entries
            float s = 0.0f;
            #pragma unroll
            for (int q = 0; q < 8; ++q) {
                const unsigned wd = rw[q];
                for (int tt = 0; tt < 8; ++tt) {
                    const int j0 = (q * 8 + tt) * 4 + 2 * h2;
                    #pragma unroll
                    for (int u = 0; u < 2; ++u) {
                        const int j = j0 + u;
                        const float z = si + dst_s[j];
                        const float lv = z >= 0.0f ? z : 0.2f * z;
                        const float v = ((wd >> (j & 31)) & 1u) ? lv : 0.0f;
                        s += expf(v - mx);
                    }
                }
            }
            s += __shfl_xor(s, 16, 32);
            const float rinv = 1.0f / s;

            // pass 3: O = P @ hp  (K=256 in 64 WMMA f32 16x16x4 steps)
            v8f acc[NT];
            #pragma unroll
            for (int nt = 0; nt < NT; ++nt)
                acc[nt] = (v8f){0.f,0.f,0.f,0.f,0.f,0.f,0.f,0.f};

            #pragma unroll
            for (int q = 0; q < 8; ++q) {
                const unsigned wd = rw[q];
                for (int tt = 0; tt < 8; ++tt) {
                    const int j0 = (q * 8 + tt) * 4 + 2 * h2;
                    v2f av;
                    {
                        const float z0 = si + dst_s[j0];
                        const float l0 = z0 >= 0.0f ? z0 : 0.2f * z0;
                        const float v0 = ((wd >> (j0 & 31)) & 1u) ? l0 : 0.0f;
                        av.x = expf(v0 - mx) * rinv;
                        const float z1 = si + dst_s[j0 + 1];
                        const float l1 = z1 >= 0.0f ? z1 : 0.2f * z1;
                        const float v1 = ((wd >> ((j0 + 1) & 31)) & 1u) ? l1 : 0.0f;
                        av.y = expf(v1 - mx) * rinv;
                    }
                    #pragma unroll
                    for (int nt = 0; nt < NT; ++nt) {
                        v2f bv;
                        bv.x = hp_s[j0 * E + nt * 16 + m];
                        bv.y = hp_s[(j0 + 1) * E + nt * 16 + m];
                        acc[nt] = __builtin_amdgcn_wmma_f32_16x16x4_f32(
                            false, av, false, bv, (short)0, acc[nt], false, false);
                    }
                }
            }

            // epilogue
            #pragma unroll
            for (int nt = 0; nt < NT; ++nt) {
                #pragma unroll
                for (int r = 0; r < 8; ++r) {
                    const int row = mt * 16 + r + 8 * h2;
                    float v = acc[nt][r] + bias[nt * 16 + m];
                    if (LAYER0) v = v > 0.0f ? v : expm1f(v);   // ELU
                    out[(gbase + row) * OUTD + h * E + nt * 16 + m] = v;
                }
            }
        }
    }
}

// ---------------------------------------------------------------------------
extern "C" void kernel_launch(void* const* d_in, const int* in_sizes, int n_in,
                              void* d_out, int out_size, void* d_ws, size_t ws_size,
                              hipStream_t stream) {
    (void)in_sizes; (void)n_in; (void)out_size; (void)ws_size;

    const float* x_emb  = (const float*)d_in[0];   // (8,4096,32)
    const float* dire   = (const float*)d_in[1];   // (8,4096,6)
    const float* w0     = (const float*)d_in[3];   // (4,32,16)
    const float* a_src0 = (const float*)d_in[4];   // (4,16,1)
    const float* a_dst0 = (const float*)d_in[5];
    const float* b0     = (const float*)d_in[6];   // (16,)
    const float* w1     = (const float*)d_in[7];   // (1,64,32)
    const float* a_src1 = (const float*)d_in[8];   // (1,32,1)
    const float* a_dst1 = (const float*)d_in[9];
    const float* b1     = (const float*)d_in[10];  // (32,)
    float* outp = (float*)d_out;                   // (8,4096,32)

    // workspace carve-up (f32 words), ~35.6 MB total
    float* xnorm = (float*)d_ws;
    float* hp0   = xnorm + 1048576;   // 8*4096*32
    float* ybuf  = hp0   + 2097152;   // 8*4096*64
    float* ynorm = ybuf  + 2097152;
    float* hp1   = ynorm + 2097152;
    float* src0  = hp1   + 1048576;   // 8*16*4*256
    float* dst0  = src0  + 131072;
    float* src1  = dst0  + 131072;    // 8*16*1*256
    float* dst1  = src1  + 32768;
    unsigned* relb = (unsigned*)(dst1 + 32768);   // 8*16*256*8 u32

    dim3 grid(NBATCH), blk(256);
    rel_kernel<<<grid, blk, 0, stream>>>(dire, relb);
    inorm_kernel<32><<<grid, blk, 0, stream>>>(x_emb, xnorm);
    gemm_wmma_kernel<32, 64, true><<<grid, blk, 0, stream>>>(xnorm, w0, hp0);
    srcdst_kernel<4, 16><<<grid, blk, 0, stream>>>(hp0, a_src0, a_dst0, src0, dst0);
    attn_kernel<4, 16, true><<<grid, blk, 0, stream>>>(hp0, src0, dst0, relb, b0, ybuf);
    inorm_kernel<64><<<grid, blk, 0, stream>>>(ybuf, ynorm);
    gemm_wmma_kernel<64, 32, false><<<grid, blk, 0, stream>>>(ynorm, w1, hp1);
    srcdst_kernel<1, 32><<<grid, blk, 0, stream>>>(hp1, a_src1, a_dst1, src1, dst1);
    attn_kernel<1, 32, false><<<grid, blk, 0, stream>>>(hp1, src1, dst1, relb, b1, outp);
}